// Decoder_43568148250645
// MI455X (gfx1250) — compile-verified
//
#include <hip/hip_runtime.h>
#include <hip/hip_bf16.h>
#include <cstddef>

// ---------------------------------------------------------------------------
// MI455X (gfx1250) attention-GRU decoder step, batch=1.
// Memory-bound: ~510 MB of fp32 weights per call -> ~22us floor @ 23.3 TB/s.
// Dominant GEMV (out_W: 50257x2048) runs on V_WMMA_F32_16X16X4_F32 with a
// column-replicated B vector (fp32-exact). One-shot weight streams use
// non-temporal loads (out_W alone is 412 MB > 192 MB L2, would thrash);
// enc (reused twice) keeps temporal policy. Fixed-order reductions only.
// ---------------------------------------------------------------------------

typedef __attribute__((ext_vector_type(2))) float v2f;
typedef __attribute__((ext_vector_type(4))) float v4f;
typedef __attribute__((ext_vector_type(8))) float v8f;

#define HH   1024
#define SS   4096
#define VV   50257
#define KK2  2048

// ---- embedding lookup + concat(x, last_context) ---------------------------
__global__ void prep_kernel(const float* __restrict__ emb, const int* __restrict__ input,
                            const float* __restrict__ lastc, float* __restrict__ rnn_in) {
    int i = blockIdx.x * blockDim.x + threadIdx.x;
    if (i < HH)            rnn_in[i] = emb[(size_t)input[0] * HH + i];
    else if (i < 2 * HH)   rnn_in[i] = lastc[i - HH];
}

// ---- y[m] = dot(W[m,0:K], x) + bias[m]; one wave per row, x staged in LDS --
// W is streamed exactly once -> non-temporal loads.
__global__ void __launch_bounds__(256)
gemv_rows_nt_kernel(const float* __restrict__ W, const float* __restrict__ x,
                    const float* __restrict__ bias, float* __restrict__ y,
                    int M, int K) {
    __shared__ float xs[KK2];
    for (int i = threadIdx.x; i < K; i += blockDim.x) xs[i] = x[i];
    __syncthreads();
    int wave = threadIdx.x >> 5;
    int lane = threadIdx.x & 31;
    int m = blockIdx.x * 8 + wave;
    if (m >= M) return;                       // wave-uniform
    const float* Wm = W + (size_t)m * K;
    float acc = 0.f;
    for (int k = lane * 4; k < K; k += 128) {
        v4f w4 = __builtin_nontemporal_load((const v4f*)(Wm + k));
        acc += w4.x * xs[k] + w4.y * xs[k + 1] + w4.z * xs[k + 2] + w4.w * xs[k + 3];
    }
    for (int off = 16; off > 0; off >>= 1) acc += __shfl_down(acc, off, 32);
    if (lane == 0) y[m] = acc + (bias ? bias[m] : 0.f);
}

// ---- same, temporal policy (enc is read again by the context kernel) ------
__global__ void __launch_bounds__(256)
gemv_rows_kernel(const float* __restrict__ W, const float* __restrict__ x,
                 const float* __restrict__ bias, float* __restrict__ y,
                 int M, int K) {
    __shared__ float xs[KK2];
    for (int i = threadIdx.x; i < K; i += blockDim.x) xs[i] = x[i];
    __syncthreads();
    int wave = threadIdx.x >> 5;
    int lane = threadIdx.x & 31;
    int m = blockIdx.x * 8 + wave;
    if (m >= M) return;
    const float* Wm = W + (size_t)m * K;
    float acc = 0.f;
    for (int k = lane * 4; k < K; k += 128) {
        v4f w4 = *(const v4f*)(Wm + k);
        acc += w4.x * xs[k] + w4.y * xs[k + 1] + w4.z * xs[k + 2] + w4.w * xs[k + 3];
    }
    for (int off = 16; off > 0; off >>= 1) acc += __shfl_down(acc, off, 32);
    if (lane == 0) y[m] = acc + (bias ? bias[m] : 0.f);
}

// ---- GRU gate math --------------------------------------------------------
__global__ void gru_combine_kernel(const float* __restrict__ gi, const float* __restrict__ gh,
                                   const float* __restrict__ hprev,
                                   float* __restrict__ hnew, float* __restrict__ hout) {
    int i = blockIdx.x * blockDim.x + threadIdx.x;
    if (i >= HH) return;
    float r = 1.f / (1.f + expf(-(gi[i]          + gh[i])));
    float z = 1.f / (1.f + expf(-(gi[HH + i]     + gh[HH + i])));
    float n = tanhf(gi[2 * HH + i] + r * gh[2 * HH + i]);
    float h = (1.f - z) * n + z * hprev[i];
    hnew[i] = h;
    hout[i] = h;
}

// ---- t[k] = sum_j A[j][k] * x[j]  (transposed GEMV, partial over j-chunks) -
__global__ void gemv_cols_partial_kernel(const float* __restrict__ A, const float* __restrict__ x,
                                         float* __restrict__ part, int J, int Kdim, int Jchunk) {
    int k  = blockIdx.x * blockDim.x + threadIdx.x;
    int j0 = blockIdx.y * Jchunk;
    int j1 = min(j0 + Jchunk, J);
    float acc = 0.f;
    for (int j = j0; j < j1; ++j) acc += A[(size_t)j * Kdim + k] * x[j];
    part[(size_t)blockIdx.y * Kdim + k] = acc;
}

__global__ void reduce_parts_kernel(const float* __restrict__ part, float* __restrict__ out,
                                    int parts, int Kdim) {
    int k = blockIdx.x * blockDim.x + threadIdx.x;
    if (k >= Kdim) return;
    float acc = 0.f;
    for (int p = 0; p < parts; ++p) acc += part[(size_t)p * Kdim + k];
    out[k] = acc;
}

// ---- softmax over 4096 energies (single block, fixed-order reductions) ----
__global__ void __launch_bounds__(1024)
softmax_kernel(const float* __restrict__ e, float* __restrict__ w1, float* __restrict__ w2, int n) {
    __shared__ float sred[32];
    int tid = threadIdx.x;
    float lmax = -3.4e38f;
    for (int i = tid; i < n; i += 1024) lmax = fmaxf(lmax, e[i]);
    for (int off = 16; off > 0; off >>= 1) lmax = fmaxf(lmax, __shfl_down(lmax, off, 32));
    if ((tid & 31) == 0) sred[tid >> 5] = lmax;
    __syncthreads();
    if (tid < 32) {
        float v = sred[tid];
        for (int off = 16; off > 0; off >>= 1) v = fmaxf(v, __shfl_down(v, off, 32));
        if (tid == 0) sred[0] = v;
    }
    __syncthreads();
    float gmax = sred[0];
    __syncthreads();
    float lsum = 0.f;
    for (int i = tid; i < n; i += 1024) lsum += expf(e[i] - gmax);
    for (int off = 16; off > 0; off >>= 1) lsum += __shfl_down(lsum, off, 32);
    if ((tid & 31) == 0) sred[tid >> 5] = lsum;
    __syncthreads();
    if (tid < 32) {
        float v = sred[tid];
        for (int off = 16; off > 0; off >>= 1) v += __shfl_down(v, off, 32);
        if (tid == 0) sred[0] = v;
    }
    __syncthreads();
    float inv = 1.f / sred[0];
    for (int i = tid; i < n; i += 1024) {
        float p = expf(e[i] - gmax) * inv;
        w1[i] = p; w2[i] = p;
    }
}

// ---- context[h] = sum_i w[i]*enc[i][h] (partials over i-chunks) -----------
__global__ void ctx_partial_kernel(const float* __restrict__ enc, const float* __restrict__ w,
                                   float* __restrict__ part, int Sn, int Ichunk) {
    int h  = blockIdx.x * blockDim.x + threadIdx.x;
    int i0 = blockIdx.y * Ichunk;
    int i1 = min(i0 + Ichunk, Sn);
    float acc = 0.f;
    for (int i = i0; i < i1; ++i) acc += w[i] * enc[(size_t)i * HH + h];
    part[(size_t)blockIdx.y * HH + h] = acc;
}

__global__ void ctx_finish_kernel(const float* __restrict__ part, int parts,
                                  const float* __restrict__ h1,
                                  float* __restrict__ ctx_out, float* __restrict__ xc) {
    int h = blockIdx.x * blockDim.x + threadIdx.x;
    if (h >= HH) return;
    float acc = 0.f;
    for (int p = 0; p < parts; ++p) acc += part[(size_t)p * HH + h];
    ctx_out[h]   = acc;
    xc[h]        = h1[h];
    xc[HH + h]   = acc;
}

// ---- logits GEMV on the f32 tensor pipe -----------------------------------
// Wave computes 16 rows of out_W. A: documented 16x4 f32 layout -> one b64
// NT load per lane (lane&15 = row, lane>>4 selects K{0,1}/{2,3}). B: the
// activation vector replicated across all 16 columns (LDS broadcast), so
// every D column holds the same 16 dot products; lanes 0/16 carry D[:,0].
__global__ void __launch_bounds__(256)
logits_wmma_kernel(const float* __restrict__ W, const float* __restrict__ xc,
                   const float* __restrict__ bias, float* __restrict__ y,
                   int Vn, int Kdim) {
    __shared__ float xs[KK2];
    for (int i = threadIdx.x; i < Kdim; i += blockDim.x) xs[i] = xc[i];
    __syncthreads();
    int wave  = threadIdx.x >> 5;
    int lane  = threadIdx.x & 31;
    int tile  = blockIdx.x * 8 + wave;
    int ntile = (Vn + 15) >> 4;
    if (tile >= ntile) return;                // wave-uniform: EXEC stays full for WMMA
    int m0  = tile * 16;
    int row = m0 + (lane & 15);
    if (row >= Vn) row = Vn - 1;              // branchless clamp, values discarded
    int koff = (lane >> 4) << 1;              // 0 for lanes 0-15, 2 for lanes 16-31
    const float* Wr = W + (size_t)row * Kdim + koff;
    v8f c = {0.f, 0.f, 0.f, 0.f, 0.f, 0.f, 0.f, 0.f};
#pragma unroll 4
    for (int kb = 0; kb < Kdim; kb += 4) {
        v2f a = __builtin_nontemporal_load((const v2f*)(Wr + kb)); // A: K=kb+koff, kb+koff+1
        v2f b;                                                     // B replicated across N
        b.x = xs[kb + koff];
        b.y = xs[kb + koff + 1];
        c = __builtin_amdgcn_wmma_f32_16x16x4_f32(
                false, a, false, b, (short)0, c, false, false);
    }
    if ((lane & 15) == 0) {                   // lane 0 -> rows m0..m0+7, lane 16 -> m0+8..m0+15
        int base = m0 + ((lane >> 4) << 3);
#pragma unroll
        for (int r = 0; r < 8; ++r) {
            int m = base + r;
            if (m < Vn) y[m] = c[r] + bias[m];
        }
    }
}

// ---- log_softmax over 50257 logits ----------------------------------------
__global__ void __launch_bounds__(1024)
log_softmax_kernel(const float* __restrict__ logits, float* __restrict__ out, int n) {
    __shared__ float sred[32];
    int tid = threadIdx.x;
    float lmax = -3.4e38f;
    for (int i = tid; i < n; i += 1024) lmax = fmaxf(lmax, logits[i]);
    for (int off = 16; off > 0; off >>= 1) lmax = fmaxf(lmax, __shfl_down(lmax, off, 32));
    if ((tid & 31) == 0) sred[tid >> 5] = lmax;
    __syncthreads();
    if (tid < 32) {
        float v = sred[tid];
        for (int off = 16; off > 0; off >>= 1) v = fmaxf(v, __shfl_down(v, off, 32));
        if (tid == 0) sred[0] = v;
    }
    __syncthreads();
    float gmax = sred[0];
    __syncthreads();
    float lsum = 0.f;
    for (int i = tid; i < n; i += 1024) lsum += expf(logits[i] - gmax);
    for (int off = 16; off > 0; off >>= 1) lsum += __shfl_down(lsum, off, 32);
    if ((tid & 31) == 0) sred[tid >> 5] = lsum;
    __syncthreads();
    if (tid < 32) {
        float v = sred[tid];
        for (int off = 16; off > 0; off >>= 1) v += __shfl_down(v, off, 32);
        if (tid == 0) sred[0] = v;
    }
    __syncthreads();
    float lse = gmax + logf(sred[0]);
    for (int i = tid; i < n; i += 1024) out[i] = logits[i] - lse;
}

// ---------------------------------------------------------------------------
extern "C" void kernel_launch(void* const* d_in, const int* in_sizes, int n_in,
                              void* d_out, int out_size, void* d_ws, size_t ws_size,
                              hipStream_t stream) {
    (void)in_sizes; (void)n_in; (void)out_size; (void)ws_size;
    const int*   input  = (const int*)  d_in[0];
    const float* lastc  = (const float*)d_in[1];
    const float* hidden = (const float*)d_in[2];   // (2,1,H)
    const float* enc    = (const float*)d_in[3];   // (S,1,H)
    const float* emb    = (const float*)d_in[4];
    const float* attn_W = (const float*)d_in[5];
    // d_in[6] = attn_b: uniform additive shift of the energies -> cancels in softmax.
    const float* Wih0 = (const float*)d_in[7];
    const float* Whh0 = (const float*)d_in[8];
    const float* bih0 = (const float*)d_in[9];
    const float* bhh0 = (const float*)d_in[10];
    const float* Wih1 = (const float*)d_in[11];
    const float* Whh1 = (const float*)d_in[12];
    const float* bih1 = (const float*)d_in[13];
    const float* bhh1 = (const float*)d_in[14];
    const float* outW = (const float*)d_in[15];
    const float* outb = (const float*)d_in[16];

    float* out_logp   = (float*)d_out;             // V
    float* out_ctx    = out_logp + VV;             // H
    float* out_hidden = out_ctx + HH;              // 2H (h0, h1)
    float* out_attnw  = out_hidden + 2 * HH;       // S

    float* w          = (float*)d_ws;
    float* ws_rnn_in  = w; w += 2 * HH;            // 2048
    float* ws_gi      = w; w += 3 * HH;            // 3072 (reused GRU0/GRU1)
    float* ws_gh      = w; w += 3 * HH;            // 3072
    float* ws_h0      = w; w += HH;
    float* ws_h1      = w; w += HH;
    float* ws_tpart   = w; w += 8 * HH;            // transposed-GEMV partials
    float* ws_t       = w; w += HH;                // attn_W^T @ h1
    float* ws_energ   = w; w += SS;
    float* ws_attnw   = w; w += SS;
    float* ws_ctxpart = w; w += 32 * HH;
    float* ws_xc      = w; w += 2 * HH;            // concat(h1, context)
    float* ws_logits  = w; w += VV;

    // 1) x = emb[token]; rnn_in = concat(x, last_context)
    prep_kernel<<<(2 * HH) / 256, 256, 0, stream>>>(emb, input, lastc, ws_rnn_in);

    // 2) GRU layer 0 (one-shot weight streams -> NT)
    gemv_rows_nt_kernel<<<3 * HH / 8, 256, 0, stream>>>(Wih0, ws_rnn_in, bih0, ws_gi, 3 * HH, 2 * HH);
    gemv_rows_nt_kernel<<<3 * HH / 8, 256, 0, stream>>>(Whh0, hidden,    bhh0, ws_gh, 3 * HH, HH);
    gru_combine_kernel<<<HH / 256, 256, 0, stream>>>(ws_gi, ws_gh, hidden, ws_h0, out_hidden);

    // 3) GRU layer 1
    gemv_rows_nt_kernel<<<3 * HH / 8, 256, 0, stream>>>(Wih1, ws_h0,       bih1, ws_gi, 3 * HH, HH);
    gemv_rows_nt_kernel<<<3 * HH / 8, 256, 0, stream>>>(Whh1, hidden + HH, bhh1, ws_gh, 3 * HH, HH);
    gru_combine_kernel<<<HH / 256, 256, 0, stream>>>(ws_gi, ws_gh, hidden + HH, ws_h1, out_hidden + HH);

    // 4) attention: energies = enc @ (attn_W^T @ h1)   [associativity rewrite]
    gemv_cols_partial_kernel<<<dim3(HH / 256, 8), 256, 0, stream>>>(attn_W, ws_h1, ws_tpart, HH, HH, 128);
    reduce_parts_kernel<<<HH / 256, 256, 0, stream>>>(ws_tpart, ws_t, 8, HH);
    gemv_rows_kernel<<<SS / 8, 256, 0, stream>>>(enc, ws_t, nullptr, ws_energ, SS, HH);  // enc reused -> temporal
    softmax_kernel<<<1, 1024, 0, stream>>>(ws_energ, ws_attnw, out_attnw, SS);

    // 5) context = attn_w @ enc ; xc = concat(h1, context)
    ctx_partial_kernel<<<dim3(HH / 256, 32), 256, 0, stream>>>(enc, ws_attnw, ws_ctxpart, SS, SS / 32);
    ctx_finish_kernel<<<HH / 256, 256, 0, stream>>>(ws_ctxpart, 32, ws_h1, out_ctx, ws_xc);

    // 6) logits = xc @ out_W^T + out_b  (WMMA f32 path, 412 MB NT stream — the roofline term)
    int ntiles = (VV + 15) / 16;                   // 3142 waves of 16 rows
    logits_wmma_kernel<<<(ntiles + 7) / 8, 256, 0, stream>>>(outW, ws_xc, outb, ws_logits, VV, 2 * HH);

    // 7) log_softmax
    log_softmax_kernel<<<1, 1024, 0, stream>>>(ws_logits, out_logp, VV);
}